// GCN_3444563771481
// MI455X (gfx1250) — compile-verified
//
#include <hip/hip_runtime.h>

// GCN 2-layer, fp32 end-to-end on MI455X (gfx1250).
// - Dense transforms: V_WMMA_F32_16X16X4_F32 (one wave32 per 16-row tile).
// - x @ W1 stages x tiles into LDS with async-to-LDS DMA (ASYNCcnt),
//   double-buffered, so the dominant 410 MB x-read is fully coalesced and
//   overlapped with WMMA compute.
// - Aggregation: hardware global_atomic_add_f32 scatter (bandwidth floor).

typedef __attribute__((ext_vector_type(2))) float v2f;
typedef __attribute__((ext_vector_type(8))) float v8f;

#define N_NODES 200000
#define N_EDGES 6400000
#define F_IN    512
#define HID     16
#define C_OUT   7
#define N_TILES (N_NODES / 16)   // 12500, exact
#define KC      64               // K-chunk staged per double-buffer step
#define NCHUNK  (F_IN / KC)      // 8
#define LROW    68               // padded LDS row stride (floats) -> no bank conflicts

// ---------------------------------------------------------------- init ------
__global__ void k_init(float* __restrict__ deg, float* __restrict__ out1,
                       float* __restrict__ out2, const float* __restrict__ b2) {
    int n = blockIdx.x * blockDim.x + threadIdx.x;
    if (n >= N_NODES) return;
    deg[n] = 1.0f;                        // self-loop contributes 1
#pragma unroll
    for (int j = 0; j < HID; ++j) out1[(size_t)n * HID + j] = 0.0f;
#pragma unroll
    for (int j = 0; j < C_OUT; ++j) out2[(size_t)n * C_OUT + j] = b2[j];
}

// ------------------------------------------------------------- degree -------
__global__ void k_degree(const int* __restrict__ dst, float* __restrict__ deg) {
    int e = blockIdx.x * blockDim.x + threadIdx.x;
    if (e >= N_EDGES) return;
    unsafeAtomicAdd(&deg[dst[e]], 1.0f);
}

__global__ void k_dinv(float* __restrict__ deg) {
    int n = blockIdx.x * blockDim.x + threadIdx.x;
    if (n >= N_NODES) return;
    deg[n] = rsqrtf(deg[n]);              // deg >= 1 always (self-loop)
}

// --------------------------------------------------- H = x @ W1 (WMMA) ------
// Block = 128 threads (4 waves) -> 4 tiles = 64 consecutive node rows.
// Async-stage x[64 rows][KC cols] into LDS (double buffered), feed WMMA A
// operands from LDS. Grid is exact (12500 tiles / 4), so EXEC is all-1s.
__global__ __launch_bounds__(128) void k_xw1(const float* __restrict__ x,
                                             const float* __restrict__ W1,
                                             float* __restrict__ H) {
    __shared__ float tile[2][64][LROW];
    const int tid  = threadIdx.x;
    const int lane = tid & 31;
    const int wave = tid >> 5;            // 0..3
    const int rc   = lane & 15;           // A row / B col within tile
    const int hi   = lane >> 4;           // 0 or 1 (half-wave)
    const int koff = hi << 1;             // 0 or 2
    const size_t rowBase = (size_t)blockIdx.x * 64;

    // Stage chunk c of all 64 rows into LDS buffer b: 8 x b128 per thread,
    // contiguous 256B per row -> fully coalesced HBM requests.
    auto issue = [&](int c, int b) {
        const float* gbase = x + rowBase * F_IN + c * KC;
#pragma unroll
        for (int i = 0; i < 8; ++i) {
            const int idx = tid + i * 128;        // 0..1023
            const int row = idx >> 4;             // 0..63
            const int seg = idx & 15;             // 16B segment within chunk
            const float* gp = gbase + (size_t)row * F_IN + seg * 4;
            const unsigned lp = (unsigned)(size_t)&tile[b][row][seg * 4];
            asm volatile("global_load_async_to_lds_b128 %0, %1, off"
                         :: "v"(lp), "v"(gp) : "memory");
        }
    };

    v8f c8 = {};
    issue(0, 0);
    for (int c = 0; c < NCHUNK; ++c) {
        const int buf = c & 1;
        if (c + 1 < NCHUNK) {
            issue(c + 1, buf ^ 1);                        // prefetch next chunk
            asm volatile("s_wait_asynccnt 0x8" ::: "memory");  // chunk c landed
        } else {
            asm volatile("s_wait_asynccnt 0x0" ::: "memory");
        }
        __syncthreads();                                  // visible to all waves
        const float* arow = &tile[buf][wave * 16 + rc][0];
        const float* wb   = W1 + (size_t)(c * KC) * HID;  // L2-resident, reused
#pragma unroll
        for (int kl = 0; kl < KC; kl += 4) {
            const int kk = kl + koff;
            v2f a, b;
            a.x = arow[kk];
            a.y = arow[kk + 1];
            b.x = wb[(size_t)kk * HID + rc];
            b.y = wb[(size_t)(kk + 1) * HID + rc];
            c8 = __builtin_amdgcn_wmma_f32_16x16x4_f32(
                     false, a, false, b, (short)0, c8, false, false);
        }
        __syncthreads();                 // all reads done before buffer reuse
    }
#pragma unroll
    for (int v = 0; v < 8; ++v) {        // D: lane<16 -> M=v, else M=v+8
        const int row = v + (hi << 3);
        H[(rowBase + wave * 16 + row) * HID + rc] = c8[v];
    }
}

// ------------------------------------------- layer-1 scatter aggregation ----
__global__ void k_agg1(const int* __restrict__ src, const int* __restrict__ dst,
                       const float* __restrict__ dinv, const float* __restrict__ H,
                       float* __restrict__ out1) {
    int e = blockIdx.x * blockDim.x + threadIdx.x;
    if (e >= N_EDGES + N_NODES) return;
    int s, d;
    if (e < N_EDGES) { s = src[e]; d = dst[e]; }
    else             { s = d = e - N_EDGES; }      // self-loops
    const float norm = dinv[s] * dinv[d];
    const float* hs = H + (size_t)s * HID;
    float* od = out1 + (size_t)d * HID;
#pragma unroll
    for (int j = 0; j < HID; ++j) unsafeAtomicAdd(&od[j], hs[j] * norm);
}

// ------------------------- G = relu(out1 + b1) @ W2 (WMMA, padded 16x16) ----
__global__ __launch_bounds__(256) void k_hw2(const float* __restrict__ out1,
                                             const float* __restrict__ b1,
                                             const float* __restrict__ W2,
                                             float* __restrict__ G) {
    const int lane = threadIdx.x & 31;
    const int wave = threadIdx.x >> 5;
    const int tile = blockIdx.x * 8 + wave;
    if (tile >= N_TILES) return;          // uniform per wave
    const int rc   = lane & 15;
    const int koff = (lane >> 4) << 1;
    const float* arow = out1 + (size_t)(tile * 16 + rc) * HID;
    v8f c = {};
#pragma unroll
    for (int k0 = 0; k0 < HID; k0 += 4) {
        const int kk = k0 + koff;
        v2f a, b;
        a.x = fmaxf(arow[kk]     + b1[kk],     0.0f);   // bias + ReLU fused
        a.y = fmaxf(arow[kk + 1] + b1[kk + 1], 0.0f);
        b.x = (rc < C_OUT) ? W2[(size_t)kk       * C_OUT + rc] : 0.0f;
        b.y = (rc < C_OUT) ? W2[(size_t)(kk + 1) * C_OUT + rc] : 0.0f;
        c = __builtin_amdgcn_wmma_f32_16x16x4_f32(
                false, a, false, b, (short)0, c, false, false);
    }
#pragma unroll
    for (int v = 0; v < 8; ++v) {
        const int row = v + ((lane >> 4) << 3);
        if (rc < 8)                                   // N x 8 padded store
            G[(size_t)(tile * 16 + row) * 8 + rc] = (rc < C_OUT) ? c[v] : 0.0f;
    }
}

// ------------------------------------------- layer-2 scatter aggregation ----
__global__ void k_agg2(const int* __restrict__ src, const int* __restrict__ dst,
                       const float* __restrict__ dinv, const float* __restrict__ G,
                       float* __restrict__ out2) {
    int e = blockIdx.x * blockDim.x + threadIdx.x;
    if (e >= N_EDGES + N_NODES) return;
    int s, d;
    if (e < N_EDGES) { s = src[e]; d = dst[e]; }
    else             { s = d = e - N_EDGES; }
    const float norm = dinv[s] * dinv[d];
    const float* gs = G + (size_t)s * 8;
    float* od = out2 + (size_t)d * C_OUT;
#pragma unroll
    for (int j = 0; j < C_OUT; ++j) unsafeAtomicAdd(&od[j], gs[j] * norm);
}

// -------------------------------------------------------------- launch ------
extern "C" void kernel_launch(void* const* d_in, const int* in_sizes, int n_in,
                              void* d_out, int out_size, void* d_ws, size_t ws_size,
                              hipStream_t stream) {
    const float* x   = (const float*)d_in[0];
    const int*   ei  = (const int*)  d_in[1];   // [2, E]: src row then dst row
    const float* W1  = (const float*)d_in[2];
    const float* b1  = (const float*)d_in[3];
    const float* W2  = (const float*)d_in[4];
    const float* b2  = (const float*)d_in[5];
    float* out = (float*)d_out;                 // [N, 7]

    const int* src = ei;
    const int* dst = ei + N_EDGES;

    // Workspace layout (floats): deg[N] | H[16N] | out1[16N] | G[8N]
    float* deg  = (float*)d_ws;
    float* H    = deg  + (size_t)N_NODES;
    float* out1 = H    + (size_t)N_NODES * HID;
    float* G    = out1 + (size_t)N_NODES * HID;

    const int TPB = 256;
    const int nb_nodes = (N_NODES + TPB - 1) / TPB;
    const int nb_edges = (N_EDGES + TPB - 1) / TPB;
    const int nb_all   = (N_EDGES + N_NODES + TPB - 1) / TPB;
    const int nb_xw1   = N_TILES / 4;           // 4 tiles per 128-thread block
    const int nb_tiles = (N_TILES + 7) / 8;     // 8 waves per block

    k_init  <<<nb_nodes, TPB, 0, stream>>>(deg, out1, out, b2);
    k_degree<<<nb_edges, TPB, 0, stream>>>(dst, deg);
    k_dinv  <<<nb_nodes, TPB, 0, stream>>>(deg);
    k_xw1   <<<nb_xw1,   128, 0, stream>>>(x, W1, H);
    k_agg1  <<<nb_all,   TPB, 0, stream>>>(src, dst, deg, H, out1);
    k_hw2   <<<nb_tiles, TPB, 0, stream>>>(out1, b1, W2, G);
    k_agg2  <<<nb_all,   TPB, 0, stream>>>(src, dst, deg, G, out);
}